// AlignmentVAE_61418032332802
// MI455X (gfx1250) — compile-verified
//
#include <hip/hip_runtime.h>
#include <hip/hip_bf16.h>

typedef float v2f __attribute__((ext_vector_type(2)));
typedef float v8f __attribute__((ext_vector_type(8)));

#define FLT_BIG 3.402823466e38f

// One wave (32 lanes) owns 16 query rows; iterates all ref points in 16-col
// tiles. d2 tile computed by V_WMMA_F32_16X16X4_F32 with
//   A row i = [x_i, y_i, n_i, 1],  B col j = [-2x_j, -2y_j, 1, n_j]
// so D[i][j] = n_i + n_j - 2*p_i.p_j  (exact fp32 squared distance).
__global__ void nn_wmma_kernel(const float2* __restrict__ qry, int nq,
                               const float2* __restrict__ ref, int nr,
                               int* __restrict__ outIdx)
{
    const int lane  = threadIdx.x & 31;
    const int gwave = (blockIdx.x * blockDim.x + threadIdx.x) >> 5;
    const int row0  = gwave * 16;
    if (row0 >= nq) return;               // uniform per wave: EXEC stays all-1s

    const int  lmod = lane & 15;
    const bool hi   = lane >= 16;

    // ---- A operand (constant over the whole scan) ----
    // lanes 0-15: VGPR0=K0(x), VGPR1=K1(y) of row row0+lane
    // lanes 16-31: VGPR0=K2(n), VGPR1=K3(1) of row row0+lane-16
    int qr = row0 + lmod; if (qr >= nq) qr = nq - 1;
    float2 qp = qry[qr];
    float  qn = qp.x * qp.x + qp.y * qp.y;
    v2f a;
    a[0] = hi ? qn   : qp.x;
    a[1] = hi ? 1.0f : qp.y;

    // Running per-lane minima: 8 rows per lane, columns j with j%16 == lmod.
    float minv[8];
    int   mini[8];
#pragma unroll
    for (int k = 0; k < 8; ++k) { minv[k] = FLT_BIG; mini[k] = 0; }

    for (int c = 0; c < nr; c += 16) {
        int jc = c + lmod; if (jc >= nr) jc = nr - 1;
        float2 rp = ref[jc];
        float  rn = rp.x * rp.x + rp.y * rp.y;
        // B operand: lanes 0-15: K0=-2x, K1=-2y ; lanes 16-31: K2=1, K3=n
        v2f b;
        b[0] = hi ? 1.0f : (-2.0f * rp.x);
        b[1] = hi ? rn   : (-2.0f * rp.y);

        v8f d = {};
        d = __builtin_amdgcn_wmma_f32_16x16x4_f32(
                /*neg_a=*/false, a, /*neg_b=*/false, b,
                /*c_mod=*/(short)0, d, /*reuse_a=*/false, /*reuse_b=*/false);

        const int j = c + lmod;           // column this lane holds
#pragma unroll
        for (int k = 0; k < 8; ++k) {
            bool lt = d[k] < minv[k];     // strict < keeps earliest index
            minv[k] = lt ? d[k] : minv[k];
            mini[k] = lt ? j    : mini[k];
        }
    }

    // ---- cross-lane argmin within each 16-lane group (lower index on tie) ----
#pragma unroll
    for (int k = 0; k < 8; ++k) {
        float v = minv[k];
        int   i = mini[k];
#pragma unroll
        for (int m = 1; m < 16; m <<= 1) {
            float ov = __shfl_xor(v, m, 32);
            int   oi = __shfl_xor(i, m, 32);
            if (ov < v || (ov == v && oi < i)) { v = ov; i = oi; }
        }
        minv[k] = v; mini[k] = i;
    }

    // lane 0 holds rows row0+0..7, lane 16 holds rows row0+8..15
    if (lmod == 0) {
        int base = row0 + (hi ? 8 : 0);
#pragma unroll
        for (int k = 0; k < 8; ++k) {
            int r = base + k;
            if (r < nq) outIdx[r] = mini[k];
        }
    }
}

// Deterministic partial sum of  sum_i (|qx-rx| + |qy-ry|)  with r = ref[idx[i]]
__global__ void err_partial_kernel(const float2* __restrict__ q,
                                   const float2* __restrict__ r,
                                   const int* __restrict__ idx, int n,
                                   float* __restrict__ partial)
{
    __shared__ float s[256];
    float acc = 0.0f;
    for (int i = blockIdx.x * 256 + threadIdx.x; i < n; i += gridDim.x * 256) {
        float2 a = q[i];
        float2 b = r[idx[i]];
        acc += fabsf(a.x - b.x) + fabsf(a.y - b.y);
    }
    s[threadIdx.x] = acc;
    __syncthreads();
    for (int off = 128; off > 0; off >>= 1) {
        if (threadIdx.x < off) s[threadIdx.x] += s[threadIdx.x + off];
        __syncthreads();
    }
    if (threadIdx.x == 0) partial[blockIdx.x] = s[0];
}

__global__ void finalize_kernel(const float* __restrict__ pA,
                                const float* __restrict__ pB, int nb,
                                float* __restrict__ out,
                                float sclA, float sclB)
{
    if (threadIdx.x == 0 && blockIdx.x == 0) {
        float sA = 0.0f, sB = 0.0f;
        for (int k = 0; k < nb; ++k) { sA += pA[k]; sB += pB[k]; }
        out[0] = sA * sclA + sB * sclB;
    }
}

extern "C" void kernel_launch(void* const* d_in, const int* in_sizes, int n_in,
                              void* d_out, int out_size, void* d_ws, size_t ws_size,
                              hipStream_t stream)
{
    const float2* pI = (const float2*)d_in[0];   // [N,2] f32
    const float2* pJ = (const float2*)d_in[1];   // [M,2] f32
    const int N = in_sizes[0] / 2;
    const int M = in_sizes[1] / 2;

    // workspace layout
    int*   idx1  = (int*)d_ws;           // N ints: for each src -> nearest dst
    int*   idx2  = idx1 + N;             // M ints: for each dst -> nearest src
    float* partA = (float*)(idx2 + M);   // 32 partials
    float* partB = partA + 32;           // 32 partials
    float* out   = (float*)d_out;

    const int THREADS = 256;             // 8 waves/block on wave32

    // pass 1: rows = pointsI, refs = pointsJ
    {
        int waves  = (N + 15) / 16;
        int blocks = (waves * 32 + THREADS - 1) / THREADS;
        nn_wmma_kernel<<<blocks, THREADS, 0, stream>>>(pI, N, pJ, M, idx1);
    }
    // pass 2: rows = pointsJ, refs = pointsI
    {
        int waves  = (M + 15) / 16;
        int blocks = (waves * 32 + THREADS - 1) / THREADS;
        nn_wmma_kernel<<<blocks, THREADS, 0, stream>>>(pJ, M, pI, N, idx2);
    }

    err_partial_kernel<<<32, THREADS, 0, stream>>>(pI, pJ, idx1, N, partA);
    err_partial_kernel<<<32, THREADS, 0, stream>>>(pJ, pI, idx2, M, partB);

    // err_i = sumA/(2N);  out = err_i*WI/N + err_j*WJ/M
    const float WI = 1.0f, WJ = 1.0f;
    const float sclA = WI / (2.0f * (float)N * (float)N);
    const float sclB = WJ / (2.0f * (float)M * (float)M);
    finalize_kernel<<<1, 32, 0, stream>>>(partA, partB, 32, out, sclA, sclB);
}